// ModuleCorefLinkerPropE2EHoi_16131897163789
// MI455X (gfx1250) — compile-verified
//
#include <hip/hip_runtime.h>
#include <hip/hip_bf16.h>

// ---------------------------------------------------------------------------
// ModuleCorefLinkerPropE2EHoi — CDNA5 (gfx1250) implementation.
// B=1, N_ALL=30000, D=768, K=512, C=16, V=300000, E=200, H=128, PROP=2.
// All GEMMs use V_WMMA_F32_16X16X4_F32 (fp32 matrix pipe, wave32).
// Each wave computes a 16x32 tile of D (two accumulators share one A frag).
// ---------------------------------------------------------------------------

#define KK   512
#define DD   768
#define HH   128
#define CC   16
#define EE   200
#define NALL 30000

typedef float v2f __attribute__((ext_vector_type(2)));
typedef float v8f __attribute__((ext_vector_type(8)));

// ---------------------------------------------------------------------------
// fp32 WMMA GEMM:  D[M,N] (+)= A[M,K] @ B[K,N]
// One wave owns a 16(M) x 32(N) tile; K stepped by 4.
// Requires: M % 16 == 0, N % 32 == 0, K % 4 == 0.
// Optional rowmap gathers A rows (entity_table[cand_ids] case).
// Fragment layouts per CDNA5 ISA (wave32):
//   A (16x4 f32): lanes 0-15 hold (k+0,k+1) of row M=lane; lanes 16-31 (k+2,k+3)
//   B (4x16 f32): lanes 0-15 rows k+0/k+1 at col=lane; lanes 16-31 rows k+2/k+3
//   C/D: vgpr v -> rows tm*16+v (lanes 0-15) / tm*16+v+8 (lanes 16-31)
// ---------------------------------------------------------------------------
__global__ void gemm_wmma_f32(const float* __restrict__ A,
                              const int* __restrict__ rowmap, int lda,
                              const float* __restrict__ B, int ldb,
                              float* __restrict__ Dm, int ldd,
                              int M, int N, int Kdim, int accumulate)
{
    const int lane   = threadIdx.x & 31;
    const int wave   = threadIdx.x >> 5;
    const int tilesN = N >> 5;                    // 32-wide N tiles
    const int tilesM = M >> 4;
    const int tile   = blockIdx.x * (blockDim.x >> 5) + wave;
    if (tile >= tilesM * tilesN) return;          // wave-uniform -> EXEC all-ones

    const int tm   = tile / tilesN;
    const int tn   = tile % tilesN;
    const int half = lane >> 4;                   // 0: lanes 0-15, 1: lanes 16-31
    const int lrow = lane & 15;

    const int mrow = tm * 16 + lrow;
    const int arow = rowmap ? rowmap[mrow] : mrow;
    const float* __restrict__ Arow = A + (long)arow * lda;
    const int ncol0 = tn * 32 + lrow;             // first 16-col subtile
    const int ncol1 = ncol0 + 16;                 // second 16-col subtile

    v8f c0 = {}, c1 = {};
    if (accumulate) {
#pragma unroll
        for (int v = 0; v < 8; ++v) {
            const long r = (long)(tm * 16 + v + half * 8) * ldd;
            c0[v] = Dm[r + ncol0];
            c1[v] = Dm[r + ncol1];
        }
    }

    for (int kk = 0; kk < Kdim; kk += 4) {
        const int k0 = kk + half * 2;
        const v2f a = *(const v2f*)(Arow + k0);   // contiguous, 8B aligned -> b64
        v2f b0, b1;
        const float* __restrict__ Bk0 = B + (long)k0 * ldb;
        const float* __restrict__ Bk1 = Bk0 + ldb;
        b0.x = Bk0[ncol0]; b0.y = Bk1[ncol0];
        b1.x = Bk0[ncol1]; b1.y = Bk1[ncol1];
        c0 = __builtin_amdgcn_wmma_f32_16x16x4_f32(false, a, false, b0,
                                                   (short)0, c0, false, false);
        c1 = __builtin_amdgcn_wmma_f32_16x16x4_f32(false, a, false, b1,
                                                   (short)0, c1, false, false);
    }

#pragma unroll
    for (int v = 0; v < 8; ++v) {
        const long r = (long)(tm * 16 + v + half * 8) * ldd;
        Dm[r + ncol0] = c0[v];
        Dm[r + ncol1] = c1[v];
    }
}

// ---------------------------------------------------------------------------
// Gathers
// ---------------------------------------------------------------------------
__global__ void gather_mentions(const float* __restrict__ cand,
                                const int* __restrict__ idx,
                                float* __restrict__ m0, float* __restrict__ m)
{
    int t = blockIdx.x * blockDim.x + threadIdx.x;
    if (t >= KK * DD) return;
    int k = t / DD, d = t - k * DD;
    float v = cand[(long)idx[k] * DD + d];
    m0[t] = v;
    m[t]  = v;
}

__global__ void gather_rowids(const int* __restrict__ candidates,
                              const int* __restrict__ idx,
                              int* __restrict__ rowid)
{
    int t = blockIdx.x * blockDim.x + threadIdx.x;
    if (t >= KK * CC) return;
    int k = t >> 4, cc = t & 15;
    rowid[t] = candidates[(long)idx[k] * CC + cc];
}

// ---------------------------------------------------------------------------
// Distance buckets: clip(floor(log2(|sb_i - sb_j| + 1)), 0, 9)
// ---------------------------------------------------------------------------
__global__ void bucket_kernel(const int* __restrict__ sb, int* __restrict__ bucket)
{
    int t = blockIdx.x * blockDim.x + threadIdx.x;
    if (t >= KK * KK) return;
    int i = t >> 9, j = t & 511;
    float dist = fabsf((float)(sb[i] - sb[j]));
    float b = floorf(log2f(dist + 1.0f));
    b = fminf(fmaxf(b, 0.0f), 9.0f);
    bucket[t] = (int)b;
}

// ---------------------------------------------------------------------------
// Coref scoring: out[i,j] = sum_h relu(hl[i,h]+hr[j,h]+dist_emb[bkt,h]+b_c[h])*v_c[h]
// withP: out = (out + p[i] + p[j]) * (i != j)
// One wave per (i,j); lanes stride H.
// ---------------------------------------------------------------------------
__global__ void coref_fused(const float* __restrict__ hl, const float* __restrict__ hr,
                            const float* __restrict__ b_c, const float* __restrict__ v_c,
                            const float* __restrict__ dist_emb,
                            const int* __restrict__ bucket,
                            const float* __restrict__ p,
                            float* __restrict__ out, int withP)
{
    int pair = blockIdx.x * (blockDim.x >> 5) + (threadIdx.x >> 5);
    if (pair >= KK * KK) return;
    int i = pair >> 9, j = pair & 511;
    int lane = threadIdx.x & 31;
    const float* HL = hl + (long)i * HH;
    const float* HR = hr + (long)j * HH;
    const float* DE = dist_emb + (long)bucket[pair] * HH;
    float acc = 0.0f;
#pragma unroll
    for (int t = 0; t < HH / 32; ++t) {
        int h = lane + t * 32;
        float x = HL[h] + HR[h] + DE[h] + b_c[h];
        acc += fmaxf(x, 0.0f) * v_c[h];
    }
#pragma unroll
    for (int off = 16; off > 0; off >>= 1)
        acc += __shfl_xor(acc, off, 32);
    if (lane == 0) {
        float s = acc;
        if (withP) s = (i == j) ? 0.0f : (s + p[i] + p[j]);
        out[pair] = s;
    }
}

// ---------------------------------------------------------------------------
// Link scoring: link[k,c] = sum_h relu(hm[k,h] + he[k,c,h] + b_l[h]) * v_l[h]
// ---------------------------------------------------------------------------
__global__ void link_fused(const float* __restrict__ hm, const float* __restrict__ he,
                           const float* __restrict__ b_l, const float* __restrict__ v_l,
                           float* __restrict__ link)
{
    int pair = blockIdx.x * (blockDim.x >> 5) + (threadIdx.x >> 5);
    if (pair >= KK * CC) return;
    int k = pair >> 4;
    int lane = threadIdx.x & 31;
    const float* HM = hm + (long)k * HH;
    const float* HE = he + (long)pair * HH;
    float acc = 0.0f;
#pragma unroll
    for (int t = 0; t < HH / 32; ++t) {
        int h = lane + t * 32;
        float x = HM[h] + HE[h] + b_l[h];
        acc += fmaxf(x, 0.0f) * v_l[h];
    }
#pragma unroll
    for (int off = 16; off > 0; off >>= 1)
        acc += __shfl_xor(acc, off, 32);
    if (lane == 0) link[pair] = acc;
}

// ---------------------------------------------------------------------------
// _add_scores: slink[k,c] = link + ss[k]; cscore[i,j] = (coref + ss[i] + ss[j])*(i!=j)
// ---------------------------------------------------------------------------
__global__ void add_scores(const float* __restrict__ link, const float* __restrict__ corefb,
                           const float* __restrict__ ss,
                           float* __restrict__ slink, float* __restrict__ cscore)
{
    int t = blockIdx.x * blockDim.x + threadIdx.x;
    if (t < KK * CC) slink[t] = link[t] + ss[t >> 4];
    if (t < KK * KK) {
        int i = t >> 9, j = t & 511;
        float v = corefb[t] + ss[i] + ss[j];
        cscore[t] = (i == j) ? 0.0f : v;
    }
}

// ---------------------------------------------------------------------------
// Row-wise masked softmax over 512 columns. One block (128 threads) per row.
// ---------------------------------------------------------------------------
__global__ void softmax_rows(const float* __restrict__ scores,
                             const float* __restrict__ mask,
                             float* __restrict__ probs)
{
    __shared__ float red[128];
    int i = blockIdx.x, t = threadIdx.x;
    const float* S = scores + (long)i * KK;
    const float* Mk = mask + (long)i * KK;
    float mx = -3.0e38f;
    for (int j = t; j < KK; j += 128) {
        float v = (Mk[j] > 0.0f) ? S[j] : -1.0e30f;
        mx = fmaxf(mx, v);
    }
    red[t] = mx; __syncthreads();
    for (int s = 64; s > 0; s >>= 1) { if (t < s) red[t] = fmaxf(red[t], red[t + s]); __syncthreads(); }
    mx = red[0]; __syncthreads();
    float sum = 0.0f;
    for (int j = t; j < KK; j += 128) {
        float v = (Mk[j] > 0.0f) ? S[j] : -1.0e30f;
        sum += expf(v - mx);
    }
    red[t] = sum; __syncthreads();
    for (int s = 64; s > 0; s >>= 1) { if (t < s) red[t] += red[t + s]; __syncthreads(); }
    float inv = 1.0f / red[0];
    for (int j = t; j < KK; j += 128) {
        float v = (Mk[j] > 0.0f) ? S[j] : -1.0e30f;
        probs[(long)i * KK + j] = expf(v - mx) * inv;
    }
}

// ---------------------------------------------------------------------------
// Gate: g = sigmoid(tmpg + bg);  m = g*m + (1-g)*ctxt
// ---------------------------------------------------------------------------
__global__ void gate_update(const float* __restrict__ tmpg, const float* __restrict__ bg,
                            const float* __restrict__ ctxt, float* __restrict__ m)
{
    int t = blockIdx.x * blockDim.x + threadIdx.x;
    if (t >= KK * DD) return;
    int d = t % DD;
    float g = 1.0f / (1.0f + expf(-(tmpg[t] + bg[d])));
    m[t] = g * m[t] + (1.0f - g) * ctxt[t];
}

// ---------------------------------------------------------------------------
// p[k] = m[k,:] . Wp + bp  (block of 256 per row)
// ---------------------------------------------------------------------------
__global__ void proj_p(const float* __restrict__ m, const float* __restrict__ Wp,
                       const float* __restrict__ bp, float* __restrict__ p)
{
    __shared__ float red[256];
    int k = blockIdx.x, t = threadIdx.x;
    float acc = 0.0f;
    for (int d = t; d < DD; d += 256) acc += m[(long)k * DD + d] * Wp[d];
    red[t] = acc; __syncthreads();
    for (int s = 128; s > 0; s >>= 1) { if (t < s) red[t] += red[t + s]; __syncthreads(); }
    if (t == 0) p[k] = red[0] + bp[0];
}

// ---------------------------------------------------------------------------
// Final assembly: scatter m into new_vecs, copy m, pack final_scores (528/row)
// ---------------------------------------------------------------------------
__global__ void finalize(const float* __restrict__ m, const int* __restrict__ idx,
                         const float* __restrict__ slink, const float* __restrict__ cscore,
                         float* __restrict__ outV, float* __restrict__ outM,
                         float* __restrict__ outS)
{
    int t = blockIdx.x * blockDim.x + threadIdx.x;
    if (t < KK * DD) {
        int k = t / DD, d = t - k * DD;
        float v = m[t];
        outM[t] = v;
        outV[(long)idx[k] * DD + d] = v;
    }
    if (t < KK * (CC + KK)) {
        int k = t / (CC + KK), c = t - k * (CC + KK);
        outS[t] = (c < CC) ? slink[k * CC + c] : cscore[((long)k << 9) + (c - CC)];
    }
}

// ---------------------------------------------------------------------------
extern "C" void kernel_launch(void* const* d_in, const int* in_sizes, int n_in,
                              void* d_out, int out_size, void* d_ws, size_t ws_size,
                              hipStream_t stream)
{
    (void)in_sizes; (void)n_in; (void)out_size; (void)ws_size;

    const float* cand        = (const float*)d_in[0];
    const float* span_scores = (const float*)d_in[1];
    const float* trimask     = (const float*)d_in[2];
    const int*   prune_idx   = (const int*)d_in[3];
    const int*   candidates  = (const int*)d_in[4];
    const int*   span_begin  = (const int*)d_in[5];
    const float* entity_tab  = (const float*)d_in[6];
    const float* Wl          = (const float*)d_in[7];
    const float* Wr          = (const float*)d_in[8];
    const float* b_c         = (const float*)d_in[9];
    const float* v_c         = (const float*)d_in[10];
    const float* dist_emb    = (const float*)d_in[11];
    const float* Wm          = (const float*)d_in[12];
    const float* We          = (const float*)d_in[13];
    const float* b_l         = (const float*)d_in[14];
    const float* v_l         = (const float*)d_in[15];
    const float* Wg          = (const float*)d_in[16];
    const float* bg          = (const float*)d_in[17];
    const float* Wp          = (const float*)d_in[18];
    const float* bp          = (const float*)d_in[19];

    // workspace carve-up (floats)
    float* ws = (float*)d_ws;
    float* m0     = ws; ws += KK * DD;
    float* m      = ws; ws += KK * DD;
    float* hl     = ws; ws += KK * HH;
    float* hr     = ws; ws += KK * HH;
    float* hm     = ws; ws += KK * HH;
    float* he     = ws; ws += KK * CC * HH;
    float* link   = ws; ws += KK * CC;
    float* corefb = ws; ws += KK * KK;
    float* probs  = ws; ws += KK * KK;
    float* ctxt   = ws; ws += KK * DD;
    float* tmpg   = ws; ws += KK * DD;
    float* pvec   = ws; ws += KK;
    float* slink  = ws; ws += KK * CC;
    float* cscore = ws; ws += KK * KK;
    int*   bucket = (int*)ws; ws += KK * KK;
    int*   rowid  = (int*)ws; ws += KK * CC;

    float* outV = (float*)d_out;                 // (1, 30000, 768)
    float* outM = outV + (long)NALL * DD;        // (1, 512, 768)
    float* outS = outM + (long)KK * DD;          // (1, 512, 528)

    const int T = 256;
    auto blk = [](long n, int t) { return (int)((n + t - 1) / t); };

    // --- gathers + buckets -------------------------------------------------
    gather_mentions<<<blk(KK * DD, T), T, 0, stream>>>(cand, prune_idx, m0, m);
    gather_rowids<<<blk(KK * CC, T), T, 0, stream>>>(candidates, prune_idx, rowid);
    bucket_kernel<<<blk(KK * KK, T), T, 0, stream>>>(span_begin, bucket);

    // --- link scores: hm = m0@Wm ; he = ent@We ; fuse ----------------------
    {
        int tiles = (KK / 16) * (HH / 32);
        gemm_wmma_f32<<<blk(tiles, 4), 128, 0, stream>>>(m0, nullptr, DD, Wm, HH, hm, HH,
                                                         KK, HH, DD, 0);
        int tilesE = ((KK * CC) / 16) * (HH / 32);
        gemm_wmma_f32<<<blk(tilesE, 4), 128, 0, stream>>>(entity_tab, rowid, EE, We, HH, he, HH,
                                                          KK * CC, HH, EE, 0);
        link_fused<<<blk(KK * CC, 8), 256, 0, stream>>>(hm, he, b_l, v_l, link);
    }

    // --- initial coref scores ---------------------------------------------
    {
        int tiles = (KK / 16) * (HH / 32);
        gemm_wmma_f32<<<blk(tiles, 4), 128, 0, stream>>>(m0, nullptr, DD, Wl, HH, hl, HH,
                                                         KK, HH, DD, 0);
        gemm_wmma_f32<<<blk(tiles, 4), 128, 0, stream>>>(m0, nullptr, DD, Wr, HH, hr, HH,
                                                         KK, HH, DD, 0);
        coref_fused<<<blk(KK * KK, 8), 256, 0, stream>>>(hl, hr, b_c, v_c, dist_emb, bucket,
                                                         nullptr, corefb, 0);
    }
    add_scores<<<blk(KK * KK, T), T, 0, stream>>>(link, corefb, span_scores, slink, cscore);

    // --- coref propagation (2 iterations) ----------------------------------
    for (int it = 0; it < 2; ++it) {
        softmax_rows<<<KK, 128, 0, stream>>>(cscore, trimask, probs);

        // ctxt = probs @ m   (512 x 512 x 768)
        {
            int tiles = (KK / 16) * (DD / 32);
            gemm_wmma_f32<<<blk(tiles, 4), 128, 0, stream>>>(probs, nullptr, KK, m, DD,
                                                             ctxt, DD, KK, DD, KK, 0);
        }
        // tmpg = m @ Wg[0:768,:] + ctxt @ Wg[768:1536,:]
        {
            int tiles = (KK / 16) * (DD / 32);
            gemm_wmma_f32<<<blk(tiles, 4), 128, 0, stream>>>(m, nullptr, DD, Wg, DD,
                                                             tmpg, DD, KK, DD, DD, 0);
            gemm_wmma_f32<<<blk(tiles, 4), 128, 0, stream>>>(ctxt, nullptr, DD,
                                                             Wg + (long)DD * DD, DD,
                                                             tmpg, DD, KK, DD, DD, 1);
        }
        gate_update<<<blk(KK * DD, T), T, 0, stream>>>(tmpg, bg, ctxt, m);

        // refreshed coref scores from updated m
        {
            int tiles = (KK / 16) * (HH / 32);
            gemm_wmma_f32<<<blk(tiles, 4), 128, 0, stream>>>(m, nullptr, DD, Wl, HH, hl, HH,
                                                             KK, HH, DD, 0);
            gemm_wmma_f32<<<blk(tiles, 4), 128, 0, stream>>>(m, nullptr, DD, Wr, HH, hr, HH,
                                                             KK, HH, DD, 0);
        }
        proj_p<<<KK, 256, 0, stream>>>(m, Wp, bp, pvec);
        coref_fused<<<blk(KK * KK, 8), 256, 0, stream>>>(hl, hr, b_c, v_c, dist_emb, bucket,
                                                         pvec, cscore, 1);
    }

    // --- outputs ------------------------------------------------------------
    hipMemcpyAsync(outV, cand, (size_t)NALL * DD * sizeof(float),
                   hipMemcpyDeviceToDevice, stream);
    finalize<<<blk(KK * DD, T), T, 0, stream>>>(m, prune_idx, slink, cscore,
                                                outV, outM, outS);
}